// Model_11888469476076
// MI455X (gfx1250) — compile-verified
//
#include <hip/hip_runtime.h>
#include <math.h>

typedef float v2f __attribute__((ext_vector_type(2)));
typedef float v8f __attribute__((ext_vector_type(8)));

#define IN_D   48
#define OUT_SP 95          // 2*48 - 1
#define NB     8
#define CIN    4
#define CMID   16
#define SEQ_LEN 10
#define LSTM_IN 32
#define HID    64
#define NOUT   128

#define XCIN   (IN_D * IN_D * IN_D)   // 110592 elements per (b, cin) plane

// ---------------------------------------------------------------------------
// One parity class of the fused ConvTranspose3d(s2,k3,p1)+bias+ReLU+spatial-sum.
// PD/PH/PW = parity of od/oh/ow. Even parity: 1 tap (k=1, input delta 0);
// odd parity: 2 taps (k=0 -> delta +1, k=2 -> delta 0). Fully unrolled.
// ---------------------------------------------------------------------------
template <int PD, int PH, int PW>
__device__ __forceinline__ void conv_class(
    const float* __restrict__ x, const float* __restrict__ w,
    int b, int part, int nparts, int m, int hi, int cin0,
    const float bb[8], float acc[8])
{
    constexpr int ND = PD ? 2 : 1, NH = PH ? 2 : 1, NW = PW ? 2 : 1;
    constexpr int NT = ND * NH * NW;
    constexpr int NOD = PD ? 47 : 48;
    constexpr int NOH = PH ? 47 : 48;
    constexpr int NCOL = PW ? 47 : 48;
    constexpr int KDT[2][2] = {{1, 1}, {0, 2}};   // kernel index per (parity, tap)
    constexpr int DLT[2][2] = {{0, 0}, {1, 0}};   // input-index delta per (parity, tap)

    // Hoisted A fragments: one v2f per tap (K = cin0, cin0+1 in .x/.y).
    v2f afrag[NT];
    {
        int ti = 0;
#pragma unroll
        for (int a_ = 0; a_ < ND; ++a_)
#pragma unroll
            for (int h_ = 0; h_ < NH; ++h_)
#pragma unroll
                for (int w_ = 0; w_ < NW; ++w_) {
                    const int wk = (KDT[PD][a_] * 3 + KDT[PH][h_]) * 3 + KDT[PW][w_];
                    afrag[ti].x = w[(cin0 * CMID + m) * 27 + wk];
                    afrag[ti].y = w[((cin0 + 1) * CMID + m) * 27 + wk];
                    ++ti;
                }
    }

    const float* __restrict__ xb = x + (long)(b * CIN + cin0) * XCIN;
    const int ntiles = NOD * NOH * 3;

    for (int tt = part; tt < ntiles; tt += nparts) {
        const int od_idx = tt / (NOH * 3);
        const int rr     = tt % (NOH * 3);
        const int oh_idx = rr / 3;
        const int colb   = (rr % 3) * 16;

        const int  col   = colb + m;
        const bool vcol  = col < NCOL;
        const int  colc  = vcol ? col : (NCOL - 1);          // clamp: address always in-bounds
        const float accmask = vcol ? 1.f : 0.f;

        const float* __restrict__ base =
            xb + ((long)od_idx * IN_D + oh_idx) * IN_D + colc;

        v8f d = {0.f, 0.f, 0.f, 0.f, 0.f, 0.f, 0.f, 0.f};
        int ti = 0;
#pragma unroll
        for (int a_ = 0; a_ < ND; ++a_)
#pragma unroll
            for (int h_ = 0; h_ < NH; ++h_)
#pragma unroll
                for (int w_ = 0; w_ < NW; ++w_) {
                    constexpr int S2 = IN_D * IN_D;
                    const int delta = (DLT[PD][a_] * IN_D + DLT[PH][h_]) * IN_D + DLT[PW][w_];
                    (void)S2;
                    v2f bv;
                    bv.x = base[delta];
                    bv.y = base[delta + XCIN];
                    d = __builtin_amdgcn_wmma_f32_16x16x4_f32(
                            false, afrag[ti], false, bv, (short)0, d, false, false);
                    ++ti;
                }

#pragma unroll
        for (int r = 0; r < 8; ++r) {
            float v = d[r] + bb[r];
            v = v > 0.f ? v : 0.f;
            acc[r] = fmaf(v, accmask, acc[r]);
        }
    }
}

// ---------------------------------------------------------------------------
// Driver kernel: blockIdx.x & 7 selects the parity class (wave-uniform),
// blockIdx.y = batch. 4 waves/block stride over the class's 16-column tiles.
// ---------------------------------------------------------------------------
__global__ __launch_bounds__(128) void convt_pool_wmma(
    const float* __restrict__ x,      // (8,4,48,48,48)
    const float* __restrict__ w,      // (4,16,3,3,3)
    const float* __restrict__ bias,   // (16)
    float* __restrict__ pooled_sum)   // (8,16) pre-zeroed accumulators
{
    const int b    = blockIdx.y;
    const int cls  = blockIdx.x & 7;
    const int lane = threadIdx.x & 31;
    const int wave = threadIdx.x >> 5;
    const int m    = lane & 15;
    const int hi   = (lane >> 4) & 1;
    const int cin0 = hi ? 2 : 0;

    const int part   = (blockIdx.x >> 3) * 4 + wave;
    const int nparts = (gridDim.x >> 3) * 4;

    __shared__ float sm[4][CMID];

    float bb[8];
#pragma unroll
    for (int r = 0; r < 8; ++r) bb[r] = bias[r + hi * 8];

    float acc[8];
#pragma unroll
    for (int r = 0; r < 8; ++r) acc[r] = 0.f;

    switch (cls) {
        case 0: conv_class<0, 0, 0>(x, w, b, part, nparts, m, hi, cin0, bb, acc); break;
        case 1: conv_class<0, 0, 1>(x, w, b, part, nparts, m, hi, cin0, bb, acc); break;
        case 2: conv_class<0, 1, 0>(x, w, b, part, nparts, m, hi, cin0, bb, acc); break;
        case 3: conv_class<0, 1, 1>(x, w, b, part, nparts, m, hi, cin0, bb, acc); break;
        case 4: conv_class<1, 0, 0>(x, w, b, part, nparts, m, hi, cin0, bb, acc); break;
        case 5: conv_class<1, 0, 1>(x, w, b, part, nparts, m, hi, cin0, bb, acc); break;
        case 6: conv_class<1, 1, 0>(x, w, b, part, nparts, m, hi, cin0, bb, acc); break;
        default: conv_class<1, 1, 1>(x, w, b, part, nparts, m, hi, cin0, bb, acc); break;
    }

    // Column reduction within each 16-lane half, 4-wave LDS combine, 1 atomic/channel.
#pragma unroll
    for (int r = 0; r < 8; ++r) {
        float v = acc[r];
        v += __shfl_xor(v, 1, 16);
        v += __shfl_xor(v, 2, 16);
        v += __shfl_xor(v, 4, 16);
        v += __shfl_xor(v, 8, 16);
        if (m == 0) sm[wave][r + hi * 8] = v;
    }
    __syncthreads();
    if (threadIdx.x < CMID) {
        float s = sm[0][threadIdx.x] + sm[1][threadIdx.x] + sm[2][threadIdx.x] + sm[3][threadIdx.x];
        atomicAdd(&pooled_sum[b * CMID + threadIdx.x], s);
    }
}

__global__ void zero_ws(float* p, int n)
{
    int i = blockIdx.x * blockDim.x + threadIdx.x;
    if (i < n) p[i] = 0.f;
}

// ---------------------------------------------------------------------------
// Tail: pooled -> projection -> 10-step LSTM(64) -> out proj. ~2 MFLOP, one block.
// ---------------------------------------------------------------------------
__global__ __launch_bounds__(256) void tail_net(
    const float* __restrict__ pooled_sum,
    const float* __restrict__ proj_w, const float* __restrict__ proj_b,
    const float* __restrict__ W_ih,  const float* __restrict__ W_hh,
    const float* __restrict__ b_ih,  const float* __restrict__ b_hh,
    const float* __restrict__ out_w, const float* __restrict__ out_b,
    float* __restrict__ out)
{
    __shared__ float pooled[NB][CMID];
    __shared__ float xs[SEQ_LEN][NB][LSTM_IN];
    __shared__ float hbuf[NB][HID];
    __shared__ float cbuf[NB][HID];
    __shared__ float gbuf[NB][4 * HID];

    const int tid = threadIdx.x;
    const float inv = 1.0f / (float)(OUT_SP * OUT_SP * OUT_SP);

    if (tid < NB * CMID) pooled[tid / CMID][tid % CMID] = pooled_sum[tid] * inv;
    __syncthreads();

    for (int idx = tid; idx < NB * SEQ_LEN * LSTM_IN; idx += 256) {
        const int b = idx / (SEQ_LEN * LSTM_IN);
        const int j = idx % (SEQ_LEN * LSTM_IN);
        float s = proj_b[j];
#pragma unroll
        for (int k = 0; k < CMID; ++k) s += pooled[b][k] * proj_w[j * CMID + k];
        const int t = j / LSTM_IN, jj = j % LSTM_IN;
        xs[t][b][jj] = s + 0.1f * pooled[b][jj & (CMID - 1)];   // tiled pooled_exp
    }
    for (int idx = tid; idx < NB * HID; idx += 256) {
        hbuf[idx / HID][idx % HID] = 0.f;
        cbuf[idx / HID][idx % HID] = 0.f;
    }
    __syncthreads();

    for (int t = 0; t < SEQ_LEN; ++t) {
        for (int idx = tid; idx < NB * 4 * HID; idx += 256) {
            const int b = idx / (4 * HID);
            const int g = idx % (4 * HID);
            float s = b_ih[g] + b_hh[g];
#pragma unroll 8
            for (int k = 0; k < LSTM_IN; ++k) s += xs[t][b][k] * W_ih[g * LSTM_IN + k];
#pragma unroll 8
            for (int k = 0; k < HID; ++k) s += hbuf[b][k] * W_hh[g * HID + k];
            gbuf[b][g] = s;
        }
        __syncthreads();
        for (int idx = tid; idx < NB * HID; idx += 256) {
            const int b = idx / HID, u = idx % HID;
            const float ig = gbuf[b][u];
            const float fg = gbuf[b][HID + u];
            const float gg = gbuf[b][2 * HID + u];
            const float og = gbuf[b][3 * HID + u];
            const float si = 1.f / (1.f + expf(-ig));
            const float sf = 1.f / (1.f + expf(-fg));
            const float so = 1.f / (1.f + expf(-og));
            const float cn = sf * cbuf[b][u] + si * tanhf(gg);
            cbuf[b][u] = cn;
            hbuf[b][u] = so * tanhf(cn);
        }
        __syncthreads();
    }

    for (int idx = tid; idx < NB * NOUT; idx += 256) {
        const int b = idx / NOUT, o = idx % NOUT;
        float s = out_b[o];
#pragma unroll 8
        for (int k = 0; k < HID; ++k) s += hbuf[b][k] * out_w[o * HID + k];
        out[idx] = s;
    }
}

// ---------------------------------------------------------------------------
extern "C" void kernel_launch(void* const* d_in, const int* in_sizes, int n_in,
                              void* d_out, int out_size, void* d_ws, size_t ws_size,
                              hipStream_t stream)
{
    const float* x       = (const float*)d_in[0];
    const float* convT_w = (const float*)d_in[1];
    const float* convT_b = (const float*)d_in[2];
    const float* proj_w  = (const float*)d_in[3];
    const float* proj_b  = (const float*)d_in[4];
    const float* W_ih    = (const float*)d_in[5];
    const float* W_hh    = (const float*)d_in[6];
    const float* b_ih    = (const float*)d_in[7];
    const float* b_hh    = (const float*)d_in[8];
    const float* out_w   = (const float*)d_in[9];
    const float* out_b   = (const float*)d_in[10];
    float* out = (float*)d_out;
    float* pooled_sum = (float*)d_ws;   // 128 floats

    zero_ws<<<1, 128, 0, stream>>>(pooled_sum, NB * CMID);
    // gridDim.x = 8 classes * 64 partitions; 4 waves/block -> 256 wave-partitions/class.
    convt_pool_wmma<<<dim3(512, NB), 128, 0, stream>>>(x, convT_w, convT_b, pooled_sum);
    tail_net<<<1, 256, 0, stream>>>(pooled_sum, proj_w, proj_b,
                                    W_ih, W_hh, b_ih, b_hh, out_w, out_b, out);
}